// GeometricPNA_8332236554740
// MI455X (gfx1250) — compile-verified
//
#include <hip/hip_runtime.h>
#include <hip/hip_bf16.h>
#include <math.h>

// ---------------- problem constants ----------------
#define NROWS 1000000
#define SI 64
#define VI 16
#define SO 128
#define VO 32
#define BSEG 4096
#define HID 32
#define XD 336           // Z_DIM(304) + H(32)
#define EPSV 1e-8f

// ---------------- workspace layout (float elements) ----------------
constexpr size_t OFF_CNT  = 0;                          // BSEG
constexpr size_t OFF_SSUM = OFF_CNT  + BSEG;            // BSEG*64
constexpr size_t OFF_SSQ  = OFF_SSUM + (size_t)BSEG*64;
constexpr size_t OFF_SMIN = OFF_SSQ  + (size_t)BSEG*64; // uint-encoded
constexpr size_t OFF_SMAX = OFF_SMIN + (size_t)BSEG*64; // uint-encoded
constexpr size_t OFF_NSUM = OFF_SMAX + (size_t)BSEG*64; // BSEG*16
constexpr size_t OFF_NSQ  = OFF_NSUM + (size_t)BSEG*16;
constexpr size_t OFF_NMIN = OFF_NSQ  + (size_t)BSEG*16; // uint-encoded
constexpr size_t OFF_NMAX = OFF_NMIN + (size_t)BSEG*16; // uint-encoded
constexpr size_t OFF_VSUM = OFF_NMAX + (size_t)BSEG*16; // BSEG*48
constexpr size_t OFF_X    = OFF_VSUM + (size_t)BSEG*48; // BSEG*336
constexpr size_t OFF_WST  = OFF_X    + (size_t)BSEG*XD; // 336*128
constexpr size_t STATS_END = OFF_X;                     // region needing init

typedef float v2f __attribute__((ext_vector_type(2)));
typedef float v8f __attribute__((ext_vector_type(8)));

// monotonic float<->uint order encoding (works for negatives)
__device__ __forceinline__ unsigned encf(float f) {
    unsigned u = __float_as_uint(f);
    return (u & 0x80000000u) ? ~u : (u | 0x80000000u);
}
__device__ __forceinline__ float decf(unsigned e) {
    return (e & 0x80000000u) ? __uint_as_float(e ^ 0x80000000u)
                             : __uint_as_float(~e);
}
__device__ __forceinline__ void fadd(float* p, float v) {
    __hip_atomic_fetch_add(p, v, __ATOMIC_RELAXED, __HIP_MEMORY_SCOPE_AGENT);
}

// ---------------- kernel 1: init accumulators ----------------
__global__ void k_init(unsigned* __restrict__ w) {
    size_t i = (size_t)blockIdx.x * blockDim.x + threadIdx.x;
    size_t stride = (size_t)gridDim.x * blockDim.x;
    for (; i < STATS_END; i += stride) {
        unsigned val = 0u;
        if ((i >= OFF_SMIN && i < OFF_SMAX) || (i >= OFF_NMIN && i < OFF_NMAX))
            val = 0xFFFFFFFFu;     // "+inf" in encoded order for atomicMin
        w[i] = val;                // 0u == "-inf" in encoded order for atomicMax
    }
}

// ---------------- kernel 2: transpose Ws[128,336] -> WsT[336,128] ----------------
__global__ void k_wst(const float* __restrict__ Ws, float* __restrict__ ws) {
    int i = blockIdx.x * 256 + threadIdx.x;
    if (i < XD * SO) {
        int k = i >> 7;      // 0..335
        int n = i & 127;     // 0..127
        ws[OFF_WST + i] = Ws[(size_t)n * XD + k];
    }
}

// ---------------- kernel 3: segmented accumulation (the 448MB pass) ----------------
__global__ void k_accum(const float* __restrict__ s, const float* __restrict__ v,
                        const int* __restrict__ idx, float* __restrict__ ws) {
    __shared__ int sidx[256];
    const int tile = blockIdx.x * 256;
    const int t = threadIdx.x;
    {
        int r = tile + t;
        sidx[t] = (r < NROWS) ? idx[r] : -1;
        if (r < NROWS)  // warm the v tile for phase B (global_prefetch_b8)
            __builtin_prefetch(v + (size_t)r * 48, 0, 1);
    }
    __syncthreads();

    unsigned* smin = (unsigned*)ws + OFF_SMIN;
    unsigned* smax = (unsigned*)ws + OFF_SMAX;
    unsigned* nmin = (unsigned*)ws + OFF_NMIN;
    unsigned* nmax = (unsigned*)ws + OFF_NMAX;

    // ---- phase A: scalar feature stats. 64 feature lanes x 4 row groups ----
    {
        const int f = t & 63, g = t >> 6;
        const int base = g * 64;
        float sum = 0.f, sq = 0.f, mn = __FLT_MAX__, mx = -__FLT_MAX__, cacc = 0.f;
        int cur = -1;
        for (int i = 0; i < 64; ++i) {
            const int li = base + i;
            const int seg = sidx[li];
            if (seg < 0) break;
            if (seg != cur) {
                if (cur >= 0) {
                    fadd(ws + OFF_SSUM + (size_t)cur * 64 + f, sum);
                    fadd(ws + OFF_SSQ  + (size_t)cur * 64 + f, sq);
                    atomicMin(smin + (size_t)cur * 64 + f, encf(mn));
                    atomicMax(smax + (size_t)cur * 64 + f, encf(mx));
                    if (f == 0) fadd(ws + OFF_CNT + cur, cacc);
                }
                cur = seg; sum = 0.f; sq = 0.f; mn = __FLT_MAX__; mx = -__FLT_MAX__; cacc = 0.f;
            }
            const float x = s[(size_t)(tile + li) * SI + f];
            sum += x; sq += x * x; mn = fminf(mn, x); mx = fmaxf(mx, x); cacc += 1.f;
        }
        if (cur >= 0) {
            fadd(ws + OFF_SSUM + (size_t)cur * 64 + f, sum);
            fadd(ws + OFF_SSQ  + (size_t)cur * 64 + f, sq);
            atomicMin(smin + (size_t)cur * 64 + f, encf(mn));
            atomicMax(smax + (size_t)cur * 64 + f, encf(mx));
            if (f == 0) fadd(ws + OFF_CNT + cur, cacc);
        }
    }

    // ---- phase B: vector channels + vnorm stats. 16 vi lanes x 16 row groups ----
    {
        const int vi = t & 15, g = t >> 4;
        const int base = g * 16;
        float v0 = 0.f, v1 = 0.f, v2 = 0.f;
        float nsum = 0.f, nsq = 0.f, nmn = __FLT_MAX__, nmx = -__FLT_MAX__;
        int cur = -1;
        for (int i = 0; i < 16; ++i) {
            const int li = base + i;
            const int seg = sidx[li];
            if (seg < 0) break;
            if (seg != cur) {
                if (cur >= 0) {
                    fadd(ws + OFF_VSUM + (size_t)cur * 48 + vi * 3 + 0, v0);
                    fadd(ws + OFF_VSUM + (size_t)cur * 48 + vi * 3 + 1, v1);
                    fadd(ws + OFF_VSUM + (size_t)cur * 48 + vi * 3 + 2, v2);
                    fadd(ws + OFF_NSUM + (size_t)cur * 16 + vi, nsum);
                    fadd(ws + OFF_NSQ  + (size_t)cur * 16 + vi, nsq);
                    atomicMin(nmin + (size_t)cur * 16 + vi, encf(nmn));
                    atomicMax(nmax + (size_t)cur * 16 + vi, encf(nmx));
                }
                cur = seg; v0 = v1 = v2 = 0.f; nsum = 0.f; nsq = 0.f;
                nmn = __FLT_MAX__; nmx = -__FLT_MAX__;
            }
            const float* vp = v + (size_t)(tile + li) * 48 + vi * 3;
            const float a = vp[0], b = vp[1], c = vp[2];
            v0 += a; v1 += b; v2 += c;
            const float n = sqrtf(fmaxf(a * a + b * b + c * c, EPSV));
            nsum += n; nsq += n * n; nmn = fminf(nmn, n); nmx = fmaxf(nmx, n);
        }
        if (cur >= 0) {
            fadd(ws + OFF_VSUM + (size_t)cur * 48 + vi * 3 + 0, v0);
            fadd(ws + OFF_VSUM + (size_t)cur * 48 + vi * 3 + 1, v1);
            fadd(ws + OFF_VSUM + (size_t)cur * 48 + vi * 3 + 2, v2);
            fadd(ws + OFF_NSUM + (size_t)cur * 16 + vi, nsum);
            fadd(ws + OFF_NSQ  + (size_t)cur * 16 + vi, nsq);
            atomicMin(nmin + (size_t)cur * 16 + vi, encf(nmn));
            atomicMax(nmax + (size_t)cur * 16 + vi, encf(nmx));
        }
    }
}

// ---------------- kernel 4: per-segment finalize + GVP vector path ----------------
__global__ void k_final(float* __restrict__ ws, const float* __restrict__ Wh,
                        const float* __restrict__ Wv, float* __restrict__ vout) {
    const int b = blockIdx.x;
    const int t = threadIdx.x;            // 128 threads
    __shared__ float cnt_s;
    __shared__ float vm[48];
    __shared__ float vh[96];
    if (t == 0) cnt_s = ws[OFF_CNT + b];
    __syncthreads();
    const float c   = cnt_s;
    const float c1  = fmaxf(c, 1.f);
    const float cm1 = fmaxf(c - 1.f, 1.f);
    const bool  ne  = c > 0.f;
    float* X = ws + OFF_X + (size_t)b * XD;
    const unsigned* smin = (const unsigned*)ws + OFF_SMIN;
    const unsigned* smax = (const unsigned*)ws + OFF_SMAX;
    const unsigned* nmin = (const unsigned*)ws + OFF_NMIN;
    const unsigned* nmax = (const unsigned*)ws + OFF_NMAX;

    if (t < 64) {
        const float sum = ws[OFF_SSUM + (size_t)b * 64 + t];
        const float sq  = ws[OFF_SSQ  + (size_t)b * 64 + t];
        const float mean = sum / c1;
        const float var  = fmaxf(sq - sum * mean, 0.f) / cm1;
        X[t]        = mean;
        X[64 + t]   = ne ? decf(smin[(size_t)b * 64 + t]) : 0.f;
        X[128 + t]  = ne ? decf(smax[(size_t)b * 64 + t]) : 0.f;
        X[192 + t]  = sqrtf(var);
    }
    if (t < 16) {
        const float sum = ws[OFF_NSUM + (size_t)b * 16 + t];
        const float sq  = ws[OFF_NSQ  + (size_t)b * 16 + t];
        const float mean = sum / c1;
        const float var  = fmaxf(sq - sum * mean, 0.f) / cm1;
        X[256 + t] = ne ? decf(nmin[(size_t)b * 16 + t]) : 0.f;
        X[272 + t] = ne ? decf(nmax[(size_t)b * 16 + t]) : 0.f;
        X[288 + t] = sqrtf(var);
    }
    if (t < 48) vm[t] = ws[OFF_VSUM + (size_t)b * 48 + t] / c1;
    __syncthreads();
    if (t < 96) {                          // vh = Wh[32,16] x vm[16,3]
        const int h = t / 3, cc = t - 3 * h;
        float acc = 0.f;
        #pragma unroll
        for (int k = 0; k < VI; ++k) acc += Wh[h * VI + k] * vm[k * 3 + cc];
        vh[t] = acc;
    }
    __syncthreads();
    if (t < 32) {                          // vn = ||vh|| (clipped)
        const float d = vh[t*3]*vh[t*3] + vh[t*3+1]*vh[t*3+1] + vh[t*3+2]*vh[t*3+2];
        X[304 + t] = sqrtf(fmaxf(d, EPSV));
    }
    if (t < 96) {                          // v_out = Wv[32,32] x vh[32,3]
        const int o = t / 3, cc = t - 3 * o;
        float acc = 0.f;
        #pragma unroll
        for (int h = 0; h < HID; ++h) acc += Wv[o * HID + h] * vh[h * 3 + cc];
        vout[(size_t)b * 96 + t] = acc;
    }
}

// ---------------- kernel 5: s_out = X[4096,336] @ WsT[336,128] + bs via f32 WMMA ----
__global__ void k_gemm(const float* __restrict__ ws, const float* __restrict__ bs,
                       float* __restrict__ out) {
    const float* X  = ws + OFF_X;
    const float* WT = ws + OFF_WST;
    const int lane = threadIdx.x & 31;
    const int wv   = threadIdx.x >> 5;          // 0..7 -> n-tile
    const int mb   = blockIdx.x << 4;           // 256 m-tiles
    const int nb   = wv << 4;
    const int l15  = lane & 15;
    const int koff = (lane >> 4) << 1;          // lanes 0-15: K {k,k+1}; 16-31: {k+2,k+3}
    const float* Xr = X  + (size_t)(mb + l15) * XD + koff;
    const float* Wc = WT + (size_t)koff * SO + nb + l15;
    v8f c = {0.f, 0.f, 0.f, 0.f, 0.f, 0.f, 0.f, 0.f};
    for (int k = 0; k < XD; k += 4) {
        v2f a, b;
        a.x = Xr[k];
        a.y = Xr[k + 1];
        b.x = Wc[(size_t)k * SO];
        b.y = Wc[(size_t)(k + 1) * SO];
        c = __builtin_amdgcn_wmma_f32_16x16x4_f32(
                /*neg_a=*/false, a, /*neg_b=*/false, b,
                /*c_mod=*/(short)0, c, /*reuse_a=*/false, /*reuse_b=*/false);
    }
    const int col = nb + l15;
    const float bias = bs[col];
    const int rbase = mb + ((lane >> 4) << 3);  // D: VGPR j -> row j (lanes<16) / j+8
    #pragma unroll
    for (int j = 0; j < 8; ++j)
        out[(size_t)(rbase + j) * SO + col] = c[j] + bias;
}

// ---------------- launcher ----------------
extern "C" void kernel_launch(void* const* d_in, const int* in_sizes, int n_in,
                              void* d_out, int out_size, void* d_ws, size_t ws_size,
                              hipStream_t stream) {
    const float* s   = (const float*)d_in[0];
    const float* v   = (const float*)d_in[1];
    const int*   idx = (const int*)d_in[2];
    // d_in[3] = batch_size (compile-time BSEG=4096)
    const float* Wh  = (const float*)d_in[4];
    const float* Ws  = (const float*)d_in[5];
    const float* bs  = (const float*)d_in[6];
    const float* Wv  = (const float*)d_in[7];
    float* out = (float*)d_out;
    float* ws  = (float*)d_ws;

    k_init<<<1024, 256, 0, stream>>>((unsigned*)d_ws);
    k_wst<<<(XD * SO + 255) / 256, 256, 0, stream>>>(Ws, ws);
    k_accum<<<(NROWS + 255) / 256, 256, 0, stream>>>(s, v, idx, ws);
    k_final<<<BSEG, 128, 0, stream>>>(ws, Wh, Wv, out + (size_t)BSEG * SO);
    k_gemm<<<BSEG / 16, 256, 0, stream>>>(ws, bs, out);
}